// InstanceSampling_72215580115026
// MI455X (gfx1250) — compile-verified
//
#include <hip/hip_runtime.h>
#include <hip/hip_bf16.h>
#include <math.h>

// Problem constants (from reference): B=2, N=8192, C=3, L = B*N = 16384
#define BATCHES   2
#define NPTS      8192
#define LTOT      16384
#define THREADS   1024
#define PER       16            // LTOT / THREADS

// ---- LDS layout (dynamic shared memory, single workgroup; 320KB/WGP on CDNA5) ----
// [0, 196608)      : phase 1-3: u64 keys[16384] (uses first 131072 bytes)
//                    phase 4-5: float pts[16384*3] (sorted, slot-major)
#define SLBL_OFF  196608        // u32[1024]  packed 2-bit labels per sorted slot (4KB)
#define SBAT_OFF  200704        // u32[512]   batch bit per sorted slot (2KB)
#define REM_OFF   202752        // u32[512]   "remaining" bitmask per sorted slot (2KB)
#define BC_OFF    204800        // broadcast scalars (64B)
#define SMEM_BYTES 204864

// ---- workspace layout in d_ws ----
// [0, 131072)       : u64 keys_ws[16384]  (sorted keys; low 32 bits = original index)
// [131072, 196608)  : i32 sel_ws[16384]   (selected sorted-slot per greedy iteration)
// [196608, 196612)  : i32 K               (number of selections)

__global__ __launch_bounds__(THREADS, 1)
void InstanceSampling_greedy_kernel(const float* __restrict__ xyz,
                                    const float* __restrict__ cls,
                                    unsigned long long* __restrict__ keys_ws,
                                    int* __restrict__ sel_ws,
                                    int* __restrict__ K_ws) {
    extern __shared__ char smem[];
    unsigned long long* keys = (unsigned long long*)smem;
    float*    pts  = (float*)smem;
    unsigned* slbl = (unsigned*)(smem + SLBL_OFF);
    unsigned* sbat = (unsigned*)(smem + SBAT_OFF);
    unsigned* rem  = (unsigned*)(smem + REM_OFF);
    float*    bcf  = (float*)(smem + BC_OFF);        // [0]=px [1]=py [2]=pz [3]=r2
    int*      bci  = (int*)(smem + BC_OFF + 16);     // [0]=done [1]=batch(cur)

    const int tid  = threadIdx.x;
    const int base = tid * PER;

    // ---------------- Phase 1: keys = (~mono(score) << 32) | idx -------------
    for (int k = 0; k < PER; ++k) {
        int o = base + k;
        float c0 = cls[o * 3 + 0];
        float c1 = cls[o * 3 + 1];
        float c2 = cls[o * 3 + 2];
        float m = c0;
        if (c1 > m) m = c1;
        if (c2 > m) m = c2;
        float score = 1.0f / (1.0f + expf(-m));
        float skey  = (score >= 0.5f) ? score : -INFINITY;  // invalid -> -inf
        unsigned fb   = __float_as_uint(skey);
        unsigned mono = (fb & 0x80000000u) ? ~fb : (fb | 0x80000000u); // ascending float order
        unsigned flip = ~mono;                                         // descending score order
        keys[o] = ((unsigned long long)flip << 32) | (unsigned)o;      // tie-break: index asc
    }

    // ---------------- Phase 2: bitonic sort (ascending u64) ------------------
    for (unsigned kk = 2; kk <= LTOT; kk <<= 1) {
        for (unsigned j = kk >> 1; j > 0; j >>= 1) {
            __syncthreads();
            #pragma unroll
            for (unsigned s = 0; s < PER; ++s) {
                unsigned idx = tid + s * THREADS;
                unsigned partner = idx ^ j;
                if (partner > idx) {
                    bool asc = ((idx & kk) == 0);
                    unsigned long long a = keys[idx];
                    unsigned long long b = keys[partner];
                    if ((a > b) == asc) { keys[idx] = b; keys[partner] = a; }
                }
            }
        }
    }
    __syncthreads();

    // ------- Phase 3: per-slot metadata + async LDS->global key dump ---------
    unsigned myorig[PER];
    unsigned chunk_b = 0, chunk_r = 0, lw = 0;
    for (int k = 0; k < PER; ++k) {
        int j = base + k;
        unsigned long long kk = keys[j];
        unsigned o    = (unsigned)kk;
        unsigned flip = (unsigned)(kk >> 32);
        myorig[k] = o;
        // valid <=> score >= 0.5  <=> flip <= ~mono(0.5f) = 0x40FFFFFF
        if (flip <= 0x40FFFFFFu) chunk_r |= (1u << k);
        if (o >> 13)             chunk_b |= (1u << k);   // batch = orig / 8192
        // label of original point (cls is L2-hot from phase 1)
        float c0 = cls[(size_t)o * 3 + 0];
        float c1 = cls[(size_t)o * 3 + 1];
        float c2 = cls[(size_t)o * 3 + 2];
        int lbl = 0; float m = c0;
        if (c1 > m) { m = c1; lbl = 1; }
        if (c2 > m) { m = c2; lbl = 2; }
        lw |= ((unsigned)lbl) << (2 * k);
    }
    slbl[tid] = lw;                                   // exclusive word per thread
    unsigned pb = __shfl_xor(chunk_b, 1);             // wave32 lane-permute combine
    unsigned pr = __shfl_xor(chunk_r, 1);
    if ((tid & 1) == 0) {
        sbat[tid >> 1] = chunk_b | (pb << 16);
        rem [tid >> 1] = chunk_r | (pr << 16);
    }
    // CDNA5 async bulk copy: sorted keys LDS -> global ws (ASYNCcnt-tracked).
    // scope:SCOPE_DEV: ack only when device-coherent (read by the next kernel
    // on other WGPs).
    {
        unsigned keysbase = (unsigned)(unsigned long long)(void*)smem;  // low 32 bits = LDS offset
        unsigned long long gbase = (unsigned long long)(void*)keys_ws;
        #pragma unroll
        for (int it = 0; it < 8; ++it) {              // 8192 x 16B = 128KB
            unsigned off = (unsigned)(it * THREADS + tid) * 16u;
            unsigned laddr = keysbase + off;
            unsigned long long gaddr = gbase + off;
            asm volatile("global_store_async_from_lds_b128 %0, %1, off scope:SCOPE_DEV"
                         :: "v"(gaddr), "v"(laddr) : "memory");
        }
        asm volatile("s_wait_asynccnt 0x0" ::: "memory");
    }
    __syncthreads();   // all waves' async reads of the key region complete

    // ------- Phase 4: gather sorted points into LDS (overwrites keys) --------
    for (int k = 0; k < PER; ++k) {
        int j = base + k;
        size_t o = (size_t)myorig[k];
        pts[j * 3 + 0] = xyz[o * 3 + 0];
        pts[j * 3 + 1] = xyz[o * 3 + 1];
        pts[j * 3 + 2] = xyz[o * 3 + 2];
    }
    __syncthreads();

    // ------- Phase 5: sequential greedy suppression (all state in LDS) -------
    int p = 0, i = 0;                                  // live only on thread 0
    for (;;) {
        if (tid == 0) {
            // first remaining sorted slot: word-skip scan (pointer is monotone)
            while (p < LTOT) {
                unsigned w = rem[p >> 5] >> (p & 31);
                if (w) { p += __builtin_ctz(w); break; }
                p = (p & ~31) + 32;
            }
            if (p >= LTOT || i >= LTOT) {
                bci[0] = 1;
            } else {
                bci[0] = 0;
                int cur = p;
                bcf[0] = pts[cur * 3 + 0];
                bcf[1] = pts[cur * 3 + 1];
                bcf[2] = pts[cur * 3 + 2];
                bci[1] = (int)((sbat[cur >> 5] >> (cur & 31)) & 1u);
                // reference quirk: radius comes from sorted slot i (loop counter)
                int lbl = (int)((slbl[i >> 4] >> ((i & 15) * 2)) & 3u);
                const float R2[3] = {0.5f * 0.5f, 0.6f * 0.6f, 0.7f * 0.7f};
                bcf[3] = R2[lbl];
                sel_ws[i] = cur;
                ++i;
            }
        }
        __syncthreads();
        if (bci[0]) break;
        float px = bcf[0], py = bcf[1], pz = bcf[2], r2 = bcf[3];
        int bcur = bci[1];
        unsigned sh   = (unsigned)(tid & 1) * 16u;
        unsigned bits = (rem [tid >> 1] >> sh) & 0xFFFFu;
        unsigned bb   = (sbat[tid >> 1] >> sh) & 0xFFFFu;
        bits &= bcur ? bb : (~bb & 0xFFFFu);           // same-batch candidates only
        unsigned clr = 0;
        while (bits) {
            int k = __builtin_ctz(bits);
            bits &= bits - 1;
            int j = base + k;
            float dx = __fsub_rn(pts[j * 3 + 0], px);
            float dy = __fsub_rn(pts[j * 3 + 1], py);
            float dz = __fsub_rn(pts[j * 3 + 2], pz);
            float d2 = __fadd_rn(__fadd_rn(__fmul_rn(dx, dx), __fmul_rn(dy, dy)),
                                 __fmul_rn(dz, dz));
            if (d2 <= r2) clr |= (1u << k);
        }
        if (clr) atomicAnd(&rem[tid >> 1], ~(clr << sh));   // ds_and_b32
        __syncthreads();
    }
    if (tid == 0) *K_ws = i;
}

// ------------- Kernel C: assemble (new_xyz, sample_idx) in parallel ----------
__global__ void InstanceSampling_finalize_kernel(const float* __restrict__ xyz,
                                                 const unsigned long long* __restrict__ keys_ws,
                                                 const int* __restrict__ sel_ws,
                                                 const int* __restrict__ K_ws,
                                                 float* __restrict__ out) {
    int i = blockIdx.x * blockDim.x + threadIdx.x;
    if (i >= LTOT) return;
    int K = *K_ws;
    int s0 = 0, s1 = 0;                                  // default index 0 per reference
    if (i < K) {
        int cur = sel_ws[i];
        unsigned orig = (unsigned)keys_ws[cur];          // low 32 bits = original flat idx
        int pidx = (int)(orig & (NPTS - 1));
        if (orig >> 13) s1 = pidx; else s0 = pidx;
    }
    // sample_idx (int values written into the float output buffer)
    out[(size_t)BATCHES * LTOT * 3 + 0 * LTOT + i] = (float)s0;
    out[(size_t)BATCHES * LTOT * 3 + 1 * LTOT + i] = (float)s1;
    // new_xyz = xyz[b, sample_idx[b, i]]
    const float* src0 = xyz + (size_t)s0 * 3;
    const float* src1 = xyz + ((size_t)NPTS + (size_t)s1) * 3;
    float* d0 = out + (size_t)i * 3;
    float* d1 = out + ((size_t)LTOT + (size_t)i) * 3;
    d0[0] = src0[0]; d0[1] = src0[1]; d0[2] = src0[2];
    d1[0] = src1[0]; d1[1] = src1[1]; d1[2] = src1[2];
}

extern "C" void kernel_launch(void* const* d_in, const int* in_sizes, int n_in,
                              void* d_out, int out_size, void* d_ws, size_t ws_size,
                              hipStream_t stream) {
    const float* xyz = (const float*)d_in[0];            // (2, 8192, 3) f32
    const float* cls = (const float*)d_in[1];            // (2, 8192, 3) f32
    unsigned long long* keys_ws = (unsigned long long*)d_ws;
    int* sel_ws = (int*)((char*)d_ws + 131072);
    int* K_ws   = (int*)((char*)d_ws + 196608);
    float* out  = (float*)d_out;                         // [new_xyz(98304) | sample_idx(32768)]

    InstanceSampling_greedy_kernel<<<1, THREADS, SMEM_BYTES, stream>>>(
        xyz, cls, keys_ws, sel_ws, K_ws);
    InstanceSampling_finalize_kernel<<<LTOT / 256, 256, 0, stream>>>(
        xyz, keys_ws, sel_ws, K_ws, out);
}